// GCNModel_80951543595843
// MI455X (gfx1250) — compile-verified
//
#include <hip/hip_runtime.h>
#include <hip/hip_bf16.h>

// ---------------------------------------------------------------------------
// GCN 3-layer propagate: x = relu(scatter(norm * (x@W)[src] -> dst)) x3
// - GEMMs on V_WMMA_F32_16X16X4_F32 (fp32 in / fp32 acc, no precision loss)
// - W (128x128 fp32, 64KB) staged into LDS via the Tensor Data Mover
//   (tensor_load_to_lds + s_wait_tensorcnt), one descriptor per workgroup
// - Edge aggregation via global_atomic_add_f32; both N*128 fp32 buffers
//   (51.2 MB each) are L2-resident on MI455X (192 MB L2)
// ---------------------------------------------------------------------------

#define KDIM 128

typedef __attribute__((ext_vector_type(2))) float v2f;
typedef __attribute__((ext_vector_type(8))) float v8f;
typedef __attribute__((ext_vector_type(4))) unsigned int u32x4;
typedef __attribute__((ext_vector_type(8))) int i32x8;
typedef __attribute__((ext_vector_type(4))) int i32x4;

// ---------------------------------------------------------------------------
// degree: deg[dst] += 1
// ---------------------------------------------------------------------------
__global__ __launch_bounds__(256) void deg_kernel(const int* __restrict__ dst,
                                                  float* __restrict__ deg, int E) {
  int e = blockIdx.x * 256 + threadIdx.x;
  if (e < E) atomicAdd(&deg[dst[e]], 1.0f);
}

// dinv = deg > 0 ? rsqrt(deg) : 0  (in place)
__global__ __launch_bounds__(256) void dinv_kernel(float* __restrict__ deg, int N) {
  int i = blockIdx.x * 256 + threadIdx.x;
  if (i < N) {
    float d = deg[i];
    deg[i] = (d > 0.0f) ? rsqrtf(d) : 0.0f;
  }
}

// ---------------------------------------------------------------------------
// TDM: DMA the 128x128 fp32 W matrix from global into LDS (one descriptor).
// D# packing per CDNA5 ISA ch.8 (group0: count/lds_addr/global_addr/type=2;
// group1: data_size=4B, tensor 128x128, tile 128x128, dim0 stride 128).
// This toolchain's builtin is the 6-arg form: (g0, g1, g2, g3, extra, cpol);
// groups 2/3 + extra are zero (disabled) for a 2-D tensor.
// ---------------------------------------------------------------------------
#if __has_builtin(__builtin_amdgcn_tensor_load_to_lds)
#define HAVE_TDM 1
__device__ __forceinline__ void tdm_load_w_to_lds(const float* W, void* lds) {
  const unsigned lds_off = (unsigned)(size_t)lds;  // flat low 32 = LDS byte offset
  const unsigned long long ga = (unsigned long long)(size_t)W;
  u32x4 g0;
  g0.x = 1u;                                     // count=1, is_restore=0, gather=0
  g0.y = lds_off;                                // lds_addr[31:0]
  g0.z = (unsigned)(ga & 0xFFFFFFFFu);           // global_addr[31:0]
  g0.w = (unsigned)((ga >> 32) & 0x1FFFFFFu)     // global_addr[56:32]
         | (2u << 30);                           // type=2 ("image")
  i32x8 g1;
  g1[0] = (2 << 16);     // data_size=2 -> 4 bytes; no multicast/iterate/pad
  g1[1] = (KDIM << 16);  // tensor_dim0[15:0] @ bits 63:48
  g1[2] = (KDIM << 16);  // tensor_dim1[15:0] @ bits 95:80
  g1[3] = (KDIM << 16);  // tile_dim0       @ bits 127:112
  g1[4] = KDIM;          // tile_dim1       @ bits 143:128 (tile_dim2=0)
  g1[5] = KDIM;          // tensor_dim0_stride[31:0]
  g1[6] = 0;
  g1[7] = 0;
  const i32x4 zero4 = {0, 0, 0, 0};
  const i32x8 zero8 = {0, 0, 0, 0, 0, 0, 0, 0};
  __builtin_amdgcn_tensor_load_to_lds(g0, g1, zero4, zero4, zero8, 0);
}
#else
#define HAVE_TDM 0
#endif

// ---------------------------------------------------------------------------
// H[N,128] = (RELU ? relu(X) : X) @ W[128,128]
// X rows r < split come from X0, rows >= split from X1 (Gu|Gi without concat;
// split=60000 is a multiple of 16). One wave per 16-row tile, W in LDS,
// 8 column tiles (full 128 cols) per wave: 8 x v8f accumulators.
// ---------------------------------------------------------------------------
template <int RELU>
__global__ __launch_bounds__(256) void gemm_wmma_f32(
    const float* __restrict__ X0, const float* __restrict__ X1, int split,
    const float* __restrict__ W, float* __restrict__ H, int nRowTiles) {
  __shared__ __align__(16) float Wlds[KDIM * KDIM];  // 64 KB

  const int tid = threadIdx.x;
#if HAVE_TDM
  if (tid < 32) {  // one wave issues the DMA and waits on TENSORcnt
    tdm_load_w_to_lds(W, (void*)&Wlds[0]);
    __builtin_amdgcn_s_wait_tensorcnt(0);
  }
#else
  for (int i = tid * 4; i < KDIM * KDIM; i += 256 * 4) {
    *(float4*)&Wlds[i] = *(const float4*)&W[i];
  }
#endif
  __syncthreads();

  const int wave = tid >> 5;
  const int lane = tid & 31;
  const int tile = blockIdx.x * 8 + wave;
  if (tile >= nRowTiles) return;

  const int m = lane & 15;            // row in tile (A) / col in tile (B,C)
  const int khalf = (lane >> 4) * 2;  // 0 or 2: K pair owned by this half-wave

  const int row = tile * 16 + m;
  const float* __restrict__ xrow =
      (row < split) ? (X0 + (size_t)row * KDIM)
                    : (X1 + (size_t)(row - split) * KDIM);

  v8f c[8];
#pragma unroll
  for (int i = 0; i < 8; ++i) c[i] = (v8f){};

  for (int k0 = 0; k0 < KDIM; k0 += 4) {
    // A fragment: A[m][k0+khalf], A[m][k0+khalf+1]
    v2f a = *(const v2f*)&xrow[k0 + khalf];
    if (RELU) {
      a.x = fmaxf(a.x, 0.0f);
      a.y = fmaxf(a.y, 0.0f);
    }
#pragma unroll
    for (int nt = 0; nt < 8; ++nt) {
      // B fragment: W[k0+khalf][n], W[k0+khalf+1][n], n = nt*16 + m
      const int n = nt * 16 + m;
      v2f b;
      b.x = Wlds[(k0 + khalf) * KDIM + n];
      b.y = Wlds[(k0 + khalf + 1) * KDIM + n];
      c[nt] = __builtin_amdgcn_wmma_f32_16x16x4_f32(
          /*neg_a=*/false, a, /*neg_b=*/false, b,
          /*c_mod=*/(short)0, c[nt], /*reuse_a=*/false, /*reuse_b=*/false);
    }
  }

  // store C: VGPR v -> row tile*16 + v + (lane/16)*8, col nt*16 + (lane%16)
  const int rbase = tile * 16 + (lane >> 4) * 8;
  const int col = lane & 15;
#pragma unroll
  for (int nt = 0; nt < 8; ++nt) {
#pragma unroll
    for (int v = 0; v < 8; ++v) {
      H[(size_t)(rbase + v) * KDIM + nt * 16 + col] = c[nt][v];
    }
  }
}

// ---------------------------------------------------------------------------
// out[dst] += dinv[src]*dinv[dst] * H[src]   (one wave per edge, float4/lane)
// Edge index + dinv values are wave-uniform: scalarize via readfirstlane so
// the index/norm loads go through SMEM instead of 32 redundant VMEM lanes.
// ---------------------------------------------------------------------------
__global__ __launch_bounds__(256) void scatter_kernel(
    const float* __restrict__ H, const int* __restrict__ src,
    const int* __restrict__ dst, const float* __restrict__ dinv,
    float* __restrict__ out, int E) {
  const int widx = __builtin_amdgcn_readfirstlane(threadIdx.x >> 5);
  const int e = blockIdx.x * 8 + widx;
  if (e >= E) return;
  const int lane = threadIdx.x & 31;
  const int s = __builtin_amdgcn_readfirstlane(src[e]);
  const int d = __builtin_amdgcn_readfirstlane(dst[e]);
  const float nrm = dinv[s] * dinv[d];
  const float4 hv = *(const float4*)(H + (size_t)s * KDIM + lane * 4);
  float* op = out + (size_t)d * KDIM + lane * 4;
  atomicAdd(op + 0, hv.x * nrm);
  atomicAdd(op + 1, hv.y * nrm);
  atomicAdd(op + 2, hv.z * nrm);
  atomicAdd(op + 3, hv.w * nrm);
}

// ---------------------------------------------------------------------------
// x = relu(x)
// ---------------------------------------------------------------------------
__global__ __launch_bounds__(256) void relu_kernel(float* __restrict__ x,
                                                   size_t n) {
  size_t i = (size_t)blockIdx.x * 256 + threadIdx.x;
  if (i < n) x[i] = fmaxf(x[i], 0.0f);
}

// ---------------------------------------------------------------------------
extern "C" void kernel_launch(void* const* d_in, const int* in_sizes, int n_in,
                              void* d_out, int out_size, void* d_ws,
                              size_t ws_size, hipStream_t stream) {
  const float* Gu = (const float*)d_in[0];
  const float* Gi = (const float*)d_in[1];
  const float* W0 = (const float*)d_in[2];
  const float* W1 = (const float*)d_in[3];
  const float* W2 = (const float*)d_in[4];
  const int* ei = (const int*)d_in[5];

  const int NU = in_sizes[0] / KDIM;
  const int NI = in_sizes[1] / KDIM;
  const int N = NU + NI;
  const int E = in_sizes[5] / 2;
  const int* src = ei;
  const int* dst = ei + E;

  float* A = (float*)d_ws;             // [N,128] aggregation ping
  float* B = A + (size_t)N * KDIM;     // [N,128] GEMM output
  float* dinv = B + (size_t)N * KDIM;  // [N] deg -> dinv (in place)
  float* out = (float*)d_out;          // [N,128] final

  const size_t xbytes = (size_t)N * KDIM * sizeof(float);
  const int nTiles = (N + 15) / 16;
  const int gBlocks = (nTiles + 7) / 8;
  const int eBlocks = (E + 7) / 8;

  // degree -> dinv
  (void)hipMemsetAsync(dinv, 0, (size_t)N * sizeof(float), stream);
  deg_kernel<<<(E + 255) / 256, 256, 0, stream>>>(dst, dinv, E);
  dinv_kernel<<<(N + 255) / 256, 256, 0, stream>>>(dinv, N);

  // ---- layer 1: h = [Gu;Gi] @ W0 ; A = scatter(h)
  gemm_wmma_f32<0><<<gBlocks, 256, 0, stream>>>(Gu, Gi, NU, W0, B, nTiles);
  (void)hipMemsetAsync(A, 0, xbytes, stream);
  scatter_kernel<<<eBlocks, 256, 0, stream>>>(B, src, dst, dinv, A, E);

  // ---- layer 2: h = relu(A) @ W1 ; A = scatter(h) (A fully read before reuse)
  gemm_wmma_f32<1><<<gBlocks, 256, 0, stream>>>(A, A, N, W1, B, nTiles);
  (void)hipMemsetAsync(A, 0, xbytes, stream);
  scatter_kernel<<<eBlocks, 256, 0, stream>>>(B, src, dst, dinv, A, E);

  // ---- layer 3: h = relu(A) @ W2 ; out = relu(scatter(h))
  gemm_wmma_f32<1><<<gBlocks, 256, 0, stream>>>(A, A, N, W2, B, nTiles);
  (void)hipMemsetAsync(out, 0, xbytes, stream);
  scatter_kernel<<<eBlocks, 256, 0, stream>>>(B, src, dst, dinv, out, E);
  relu_kernel<<<(int)(((size_t)N * KDIM + 255) / 256), 256, 0, stream>>>(
      out, (size_t)N * KDIM);
}